// my_Net_63316407878362
// MI455X (gfx1250) — compile-verified
//
#include <hip/hip_runtime.h>
#include <math.h>

#define UEn 50
#define Kn  10
#define Mn  10
#define TPB 64   // 2 waves of 32 on gfx1250

typedef int v4i __attribute__((vector_size(16)));
typedef __attribute__((address_space(3))) v4i lds_v4i;

__device__ __forceinline__ float rcpf_(float x){
#if defined(__HIP_DEVICE_COMPILE__)
  return __builtin_amdgcn_rcpf(x);      // v_rcp_f32
#else
  return 1.0f / x;
#endif
}
__device__ __forceinline__ float sqrtf_(float x){
#if defined(__HIP_DEVICE_COMPILE__)
  return __builtin_amdgcn_sqrtf(x);     // v_sqrt_f32
#else
  return sqrtf(x);
#endif
}

// monotonic float<->uint mapping (total order incl. negatives)
__device__ __forceinline__ unsigned f2ord(float f){
  unsigned u = __float_as_uint(f);
  return (u & 0x80000000u) ? ~u : (u | 0x80000000u);
}
__device__ __forceinline__ float ord2f(unsigned u){
  return (u & 0x80000000u) ? __uint_as_float(u & 0x7fffffffu) : __uint_as_float(~u);
}

__global__ void init_minmax(unsigned* ws){
  if (threadIdx.x == 0){ ws[0] = 0xFFFFFFFFu; ws[1] = 0u; }
}

__global__ void reduce_minmax(const float* __restrict__ x, int n, unsigned* ws){
  int i      = blockIdx.x * blockDim.x + threadIdx.x;
  int stride = gridDim.x * blockDim.x;
  unsigned mn = 0xFFFFFFFFu, mx = 0u;
  for (; i < n; i += stride){
    unsigned o = f2ord(x[i]);
    mn = (o < mn) ? o : mn;
    mx = (o > mx) ? o : mx;
  }
  #pragma unroll
  for (int off = 16; off > 0; off >>= 1){
    unsigned a = (unsigned)__shfl_xor((int)mn, off, 32);
    unsigned b = (unsigned)__shfl_xor((int)mx, off, 32);
    mn = (a < mn) ? a : mn;
    mx = (b > mx) ? b : mx;
  }
  if ((threadIdx.x & 31) == 0){
    atomicMin(&ws[0], mn);
    atomicMax(&ws[1], mx);
  }
}

// One batch-row per lane. LDS pre-table makes the 100-iteration sigma loop
// 2 FMA + 2 v_rcp + 4 FMA/MUL + 1 ds_load_b128 per UE element.
__global__ __launch_bounds__(TPB) void solver(
    const float* __restrict__ x,     const float* __restrict__ beta,
    const float* __restrict__ alpha, const float* __restrict__ gamma,
    const float* __restrict__ mu,    const unsigned* __restrict__ mm,
    float* __restrict__ out, int B)
{
  __shared__ float4 pre[UEn * TPB];   // 51200 B: per-lane per-UE {c*MEC/UE, q*MEC/UE, lam*(c+q), w*lamqc^2}
  __shared__ float  xs [UEn * TPB];   // 12800 B: staged x tile (later reused as p staging)

  const int       tid   = threadIdx.x;
  const int       n     = B * UEn;
  const long long gbase = (long long)blockIdx.x * (UEn * TPB);

  // ---- stage this block's x tile (64 rows x 50 f32) coalesced via async copy ----
  #pragma unroll 1
  for (int i = 0; i < UEn; ++i){
    int       c = i * TPB + tid;          // float4 chunk id within tile
    long long g = gbase + (long long)c * 4;
    if (g < n){
#if __has_builtin(__builtin_amdgcn_global_load_async_to_lds_b128)
      __builtin_amdgcn_global_load_async_to_lds_b128(
          (v4i*)(x + g), (lds_v4i*)(&xs[c * 4]), 0, 0);
#else
      *(float4*)(&xs[c * 4]) = *(const float4*)(x + g);
#endif
    }
  }
#if __has_builtin(__builtin_amdgcn_global_load_async_to_lds_b128)
  asm volatile("s_wait_asynccnt 0" ::: "memory");
#endif
  __syncthreads();

  // ---- transform x in place for my private row:  t = 0.005 + scale*(x - xmin) ----
  const float xmin  = ord2f(mm[0]);
  const float xmax  = ord2f(mm[1]);
  const float scale = 0.005f * rcpf_(xmax - xmin);
  float* myx = &xs[tid * UEn];
  #pragma unroll
  for (int ue = 0; ue < UEn; ++ue)
    myx[ue] = fmaf(scale, myx[ue] - xmin, 0.005f);

  // ---- solver state in VGPRs ----
  float p[UEn];
  #pragma unroll
  for (int ue = 0; ue < UEn; ++ue) p[ue] = 0.2f;
  float sigma  = 0.5f;
  float gm     = 1.0f;            // gamma^m
  const float gamma0 = gamma[0];

  #pragma unroll 1
  for (int m = 0; m < Mn; ++m){
    // ---- phase A: per-UE constants for this m (p fixed inside k-loop) ----
    #pragma unroll
    for (int ue = 0; ue < UEn; ++ue){
      float muv   = mu[ue];                                  // wave-uniform -> s_load
      float pv    = p[ue];
      float c     = (100.0f * 1.0e6f) * muv * pv;            // N*mu*CON*p
      float q     = (1.0f - __powf(1.0f - pv, 100.0f)) * 1.0e4f;  // (1-(1-p)^N)/TAU
      float lam   = 1000.0f + (4000.0f / 49.0f) * (float)ue;
      float lamqc = lam * (c + q);
      float wv    = lam * (1.0f / 150000.0f);                // LAMDA/sum(LAMDA)
      float4 pr;
      pr.x = c * (1.0e4f / 50.0f);                           // c*MEC/UE
      pr.y = q * (1.0e4f / 50.0f);                           // q*MEC/UE
      pr.z = lamqc;
      pr.w = wv * lamqc * lamqc;                             // w * numer
      pre[ue * TPB + tid] = pr;                              // ds_store_b128 (lane-private)
    }

    // ---- phase B: 10 sigma fixed-point iterations ----
    #pragma unroll 1
    for (int k = 0; k < Kn; ++k){
      float bk  = beta[k];
      float oms = 1.0f - sigma;
      float sa = 0.0f, ga = 0.0f;
      #pragma unroll
      for (int ue = 0; ue < UEn; ++ue){
        float4 pr = pre[ue * TPB + tid];                     // ds_load_b128
        float i1  = fmaf(pr.x, oms, pr.z);                   // c*MEC*(1-s)/UE + lam(q+c)
        float i2  = fmaf(pr.y, oms, pr.z);
        float r1  = rcpf_(i1), r2 = rcpf_(i2);
        float t12 = r1 * r2;
        float wnt = pr.w * t12;                              // w*numer/(i1*i2)
        sa += wnt;
        ga  = fmaf(wnt, fmaf(pr.y, r2, pr.x * r1), ga);      // wn*r1*r2*(i1b*r1 + i2b*r2)
      }
      sigma = sigma - bk * (sa - sigma) * rcpf_(ga - 1.0f);
      sigma = fminf(fmaxf(sigma, 0.0f), 1.0f);
    }

    // ---- phase C: dual-number f_fun (value + d/dp) and p update ----
    const float V  = 1.0e4f * (1.0f - sigma);
    const float am = alpha[m];
    #pragma unroll
    for (int ue = 0; ue < UEn; ++ue){
      float lam = 1000.0f + (4000.0f / 49.0f) * (float)ue;
      float muv = mu[ue];
      float t   = myx[ue];
      float pv  = p[ue];

      float a  = (100.0f * 1.0e6f) * muv;   // dC/dp
      float b  = 1.0e6f;                    // dQ/dp = N/TAU
      float C  = a * pv, Q = b * pv;
      float d  = C - Q,  u = C + Q;
      float dd = a - b,  ud = a + b;

      float disc2 = fmaf(d, d, fmaf(2.0f * u, lam, lam * lam));
      float disc  = sqrtf_(disc2);
      float rdisc = rcpf_(disc);
      float discd = fmaf(d, dd, lam * ud) * rdisc;

      float s1  = 0.5f * (u - lam - disc), s1d = 0.5f * (ud - discd);
      float s2  = 0.5f * (u - lam + disc), s2d = 0.5f * (ud + discd);

      float P  = s1 * s2 * rdisc;
      float Pd = (fmaf(s1d, s2, s1 * s2d) - P * discd) * rdisc;

      float A1 = V - s1, A2 = V - s2;
      float rA1 = rcpf_(A1), rA2 = rcpf_(A2);

      float e0 = __expf(-V * t);                             // v_exp_f32
      float term1  = (rA1 - rA2) * e0;
      float term1d = (s1d * rA1 * rA1 - s2d * rA2 * rA2) * e0;

      float e2  = __expf(-s2 * t);
      float h2  = s2 * A2;
      float rh2 = rcpf_(h2);
      float h2d = s2d * (V - 2.0f * s2);
      float term2  =  V * e2 * rh2;
      float term2d = -V * e2 * fmaf(s2d * t, h2, h2d) * rh2 * rh2;

      float e1  = __expf(-s1 * t);
      float h1  = s1 * A1;
      float rh1 = rcpf_(h1);
      float h1d = s1d * (V - 2.0f * s1);
      float term3  = -V * e1 * rh1;
      float term3d =  V * e1 * fmaf(s1d * t, h1, h1d) * rh1 * rh1;

      float E  = term1 + term2 + term3;
      float Ed = term1d + term2d + term3d;
      float z1 = fmaf(P, E, 1.0f);
      float z2 = fmaf(Pd, E, P * Ed);

      float dG = fmaf(z2, rcpf_(1.0f - z1 - 0.01f), gm);
      pv = fmaf(-am, dG, pv);
      pv = fminf(pv, 0.3f);                                  // THRESH
      float p0 = lam * fmaf(muv, 100.0f, 1.0f) * rcpf_(1.0e8f * muv);
      pv = fmaxf(pv, p0);
      p[ue] = pv;
    }
    gm *= gamma0;
  }

  // ---- stage p into LDS and store coalesced via async LDS->global ----
  #pragma unroll
  for (int ue = 0; ue < UEn; ++ue) myx[ue] = p[ue];
  __syncthreads();
  #pragma unroll 1
  for (int i = 0; i < UEn; ++i){
    int       c = i * TPB + tid;
    long long g = gbase + (long long)c * 4;
    if (g < n){
#if __has_builtin(__builtin_amdgcn_global_store_async_from_lds_b128)
      __builtin_amdgcn_global_store_async_from_lds_b128(
          (v4i*)(out + g), (lds_v4i*)(&xs[c * 4]), 0, 0);
#else
      *(float4*)(out + g) = *(const float4*)(&xs[c * 4]);
#endif
    }
  }
#if __has_builtin(__builtin_amdgcn_global_store_async_from_lds_b128)
  asm volatile("s_wait_asynccnt 0" ::: "memory");
#endif
}

extern "C" void kernel_launch(void* const* d_in, const int* in_sizes, int n_in,
                              void* d_out, int out_size, void* d_ws, size_t ws_size,
                              hipStream_t stream){
  const float* x     = (const float*)d_in[0];
  const float* beta  = (const float*)d_in[1];
  const float* alpha = (const float*)d_in[2];
  const float* gamma = (const float*)d_in[3];
  const float* mu    = (const float*)d_in[4];
  int n = in_sizes[0];
  int B = n / UEn;
  unsigned* mm = (unsigned*)d_ws;

  init_minmax<<<1, 64, 0, stream>>>(mm);
  reduce_minmax<<<512, 256, 0, stream>>>(x, n, mm);
  int blocks = (B + TPB - 1) / TPB;
  solver<<<blocks, TPB, 0, stream>>>(x, beta, alpha, gamma, mu, mm, (float*)d_out, B);
}